// GAT_15874199126039
// MI455X (gfx1250) — compile-verified
//
#include <hip/hip_runtime.h>

#define NN 50000
#define FF 128
#define DD 64
#define EE 800000
#define GG 64
#define NEG_SLOPE 0.2f
#define EPSV 1e-16f

typedef __attribute__((ext_vector_type(2))) float v2f;
typedef __attribute__((ext_vector_type(8))) float v8f;

// ---- monotone uint encoding of float so unsigned atomicMax == float max ----
__device__ __forceinline__ unsigned f2mono(float f) {
  unsigned u = __float_as_uint(f);
  return (u & 0x80000000u) ? ~u : (u | 0x80000000u);
}
__device__ __forceinline__ float mono2f(unsigned u) {
  return (u & 0x80000000u) ? __uint_as_float(u & 0x7fffffffu)
                           : __uint_as_float(~u);
}

// ---------------- fills ----------------
__global__ void fill_f32(float* p, float v, int n) {
  int i = blockIdx.x * blockDim.x + threadIdx.x;
  if (i < n) p[i] = v;
}
__global__ void fill_u32(unsigned* p, unsigned v, int n) {
  int i = blockIdx.x * blockDim.x + threadIdx.x;
  if (i < n) p[i] = v;
}

// ---------------- fp32 WMMA GEMM: C[M x 64] = A[M x K] @ B[K x 64] ----------
// Block = 128 threads = 4 waves. B is staged once into LDS (transposed, padded
// stride) so the inner loop is: one global_load_b64 of A + 4 ds_load_b64 of B
// feeding 4 chained V_WMMA_F32_16X16X4_F32. Each wave produces a full
// 16-row x 64-col stripe, so A is read exactly once.
__global__ __launch_bounds__(128) void gemm_wmma_rows(
    const float* __restrict__ A, const float* __restrict__ B,
    float* __restrict__ C, int M, int K) {
  __shared__ float Bt[DD * (FF + 2)];         // transposed weights, 33.3 KB max
  const int ldb = K + 2;                      // pad: bank = (2n + k) mod 64
  int tid = threadIdx.x;
  for (int idx = tid; idx < K * DD; idx += 128) {
    int k = idx >> 6;                         // row in B (K x 64)
    int n = idx & 63;                         // col in B
    Bt[n * ldb + k] = B[idx];
  }
  __syncthreads();

  int wave = tid >> 5;
  int lane = tid & 31;
  int rt = blockIdx.x * 4 + wave;
  if (rt >= (M >> 4)) return;                 // wave-uniform: EXEC all-ones

  int mrow  = rt * 16 + (lane & 15);
  int khalf = (lane >> 4) << 1;               // 0 or 2 (A/B fragment K-half)
  int nlow  = lane & 15;

  v8f a0 = {0.f,0.f,0.f,0.f,0.f,0.f,0.f,0.f};
  v8f a1 = a0, a2 = a0, a3 = a0;
  const float* arow = A + (size_t)mrow * K + khalf;
  const float* bk   = Bt + khalf;
  for (int k = 0; k < K; k += 4) {
    v2f av = *(const v2f*)(arow + k);         // 8B aligned (K even)
    const float* bp = bk + k;
    v2f b0 = *(const v2f*)(bp + (nlow     ) * ldb);
    v2f b1 = *(const v2f*)(bp + (nlow + 16) * ldb);
    v2f b2 = *(const v2f*)(bp + (nlow + 32) * ldb);
    v2f b3 = *(const v2f*)(bp + (nlow + 48) * ldb);
    a0 = __builtin_amdgcn_wmma_f32_16x16x4_f32(false, av, false, b0, (short)0, a0, false, false);
    a1 = __builtin_amdgcn_wmma_f32_16x16x4_f32(false, av, false, b1, (short)0, a1, false, false);
    a2 = __builtin_amdgcn_wmma_f32_16x16x4_f32(false, av, false, b2, (short)0, a2, false, false);
    a3 = __builtin_amdgcn_wmma_f32_16x16x4_f32(false, av, false, b3, (short)0, a3, false, false);
  }
  // C/D layout: VGPR r -> M = r + 8*(lane>=16), N = lane&15 (+16 per tile)
  int rowbase = rt * 16 + ((lane >> 4) << 3);
#pragma unroll
  for (int r = 0; r < 8; ++r) {
    float* crow = C + (size_t)(rowbase + r) * DD + nlow;
    crow[0]  = a0[r];
    crow[16] = a1[r];
    crow[32] = a2[r];
    crow[48] = a3[r];
  }
}

// ---------------- per-row dot with D=64 attention vector ----------------
__global__ void rowdot64(const float* __restrict__ X, const float* __restrict__ a,
                         float* __restrict__ out, int n) {
  int i = blockIdx.x * blockDim.x + threadIdx.x;
  if (i >= n) return;
  const float4* xr = (const float4*)(X + (size_t)i * DD);
  const float4* av = (const float4*)a;
  float s = 0.f;
#pragma unroll
  for (int j = 0; j < DD / 4; ++j) {
    float4 xv = xr[j], aw = av[j];
    s += xv.x * aw.x + xv.y * aw.y + xv.z * aw.z + xv.w * aw.w;
  }
  out[i] = s;
}

// ---------------- edge phase ----------------
__global__ void edge_logits(const float* __restrict__ es, const float* __restrict__ ed,
                            const int* __restrict__ src, const int* __restrict__ dst,
                            float* __restrict__ ebuf, unsigned* __restrict__ mbuf, int n) {
  int i = blockIdx.x * blockDim.x + threadIdx.x;
  if (i >= n) return;
  int s = src[i], t = dst[i];
  float v = es[s] + ed[t];
  v = (v > 0.f) ? v : NEG_SLOPE * v;          // leaky_relu
  ebuf[i] = v;
  atomicMax(&mbuf[t], f2mono(v));             // global_atomic_max_u32
}

__global__ void edge_expsum(float* __restrict__ ebuf, const unsigned* __restrict__ mbuf,
                            const int* __restrict__ dst, float* __restrict__ z, int n) {
  int i = blockIdx.x * blockDim.x + threadIdx.x;
  if (i >= n) return;
  int t = dst[i];
  float m = mono2f(mbuf[t]);
  float p = __expf(ebuf[i] - m);
  ebuf[i] = p;                                 // logits -> numerators in place
  atomicAdd(&z[t], p);
}

// 16 threads per edge, each handles 4 dims (float4 gather, 4 f32 atomics)
__global__ void edge_scatter(const float* __restrict__ p, const float* __restrict__ z,
                             const int* __restrict__ src, const int* __restrict__ dst,
                             const float* __restrict__ xs, float* __restrict__ acc,
                             int nthreads) {
  int t = blockIdx.x * blockDim.x + threadIdx.x;
  if (t >= nthreads) return;
  int e = t >> 4, q = t & 15;
  int s = src[e], d = dst[e];
  float alpha = p[e] / (z[d] + EPSV);
  float4 v = *(const float4*)(xs + (size_t)s * DD + q * 4);
  float* ap = acc + (size_t)d * DD + q * 4;
  atomicAdd(ap + 0, alpha * v.x);
  atomicAdd(ap + 1, alpha * v.y);
  atomicAdd(ap + 2, alpha * v.z);
  atomicAdd(ap + 3, alpha * v.w);
}

__global__ void node_finalize(const float* __restrict__ acc, const float* __restrict__ bias,
                              float* __restrict__ h, int n, int do_relu) {
  int i = blockIdx.x * blockDim.x + threadIdx.x;
  if (i >= n) return;
  float v = acc[i] + bias[i & (DD - 1)];
  if (do_relu) v = fmaxf(v, 0.f);
  h[i] = v;
}

// ---------------- pooling + final linear ----------------
__global__ void pool_cnt(const int* __restrict__ batch, float* __restrict__ gcnt, int n) {
  int i = blockIdx.x * blockDim.x + threadIdx.x;
  if (i < n) atomicAdd(&gcnt[batch[i]], 1.0f);
}
__global__ void pool_sum(const float* __restrict__ h, const int* __restrict__ batch,
                         float* __restrict__ gsum, int n) {
  int i = blockIdx.x * blockDim.x + threadIdx.x;
  if (i >= n) return;
  int node = i >> 6;                           // DD == 64
  atomicAdd(&gsum[(size_t)batch[node] * DD + (i & 63)], h[i]);
}

__global__ void final_linear(const float* __restrict__ gsum, const float* __restrict__ gcnt,
                             const float* __restrict__ post, const float* __restrict__ lw,
                             const float* __restrict__ lb, float* __restrict__ out) {
  int t = threadIdx.x;                         // 256 threads = 128 rows x 2 cols
  int r = t >> 1, c = t & 1;
  float s = 0.f;
  if (r < GG) {
    float inv = 1.0f / fmaxf(gcnt[r], 1.0f);
    const float* gr = gsum + (size_t)r * DD;
#pragma unroll 8
    for (int d = 0; d < DD; ++d) s += gr[d] * inv * lw[d * 2 + c];
  } else {
    const float* pr = post + (size_t)(r - GG) * DD;
#pragma unroll 8
    for (int d = 0; d < DD; ++d) s += pr[d] * lw[d * 2 + c];
  }
  out[r * 2 + c] = s + lb[c];
}

// ---------------- driver ----------------
extern "C" void kernel_launch(void* const* d_in, const int* in_sizes, int n_in,
                              void* d_out, int out_size, void* d_ws, size_t ws_size,
                              hipStream_t stream) {
  (void)in_sizes; (void)n_in; (void)out_size; (void)ws_size;
  const float* x     = (const float*)d_in[0];
  const int*   ei    = (const int*)d_in[1];
  const int*   batch = (const int*)d_in[2];
  const float* post  = (const float*)d_in[3];
  const float* Ws[3] = {(const float*)d_in[4],  (const float*)d_in[9],  (const float*)d_in[14]};
  const float* Wd[3] = {(const float*)d_in[5],  (const float*)d_in[10], (const float*)d_in[15]};
  const float* As[3] = {(const float*)d_in[6],  (const float*)d_in[11], (const float*)d_in[16]};
  const float* Ad[3] = {(const float*)d_in[7],  (const float*)d_in[12], (const float*)d_in[17]};
  const float* Bs[3] = {(const float*)d_in[8],  (const float*)d_in[13], (const float*)d_in[18]};
  const float* lw = (const float*)d_in[19];
  const float* lb = (const float*)d_in[20];
  const int* src = ei;
  const int* dst = ei + EE;

  // workspace carve-out (~56 MB of floats)
  float*    ws   = (float*)d_ws;
  float*    h    = ws;                          // N*D
  float*    xs   = h   + (size_t)NN * DD;       // N*D
  float*    xd   = xs  + (size_t)NN * DD;       // N*D
  float*    acc  = xd  + (size_t)NN * DD;       // N*D
  float*    es   = acc + (size_t)NN * DD;       // N
  float*    edv  = es  + NN;                    // N
  float*    zbuf = edv + NN;                    // N
  unsigned* mbuf = (unsigned*)(zbuf + NN);      // N
  float*    ebuf = (float*)(mbuf + NN);         // E
  float*    gsum = ebuf + EE;                   // G*D
  float*    gcnt = gsum + (size_t)GG * DD;      // G

  const int T = 256;
  auto nb = [](long n, int t) { return (int)((n + t - 1) / t); };

  const float* Hin = x;
  int K = FF;
  for (int l = 0; l < 3; ++l) {
    int rowTiles = NN / 16;                     // 3125
    int gblocks = nb(rowTiles, 4);              // 4 row-stripe waves / block
    gemm_wmma_rows<<<gblocks, 128, 0, stream>>>(Hin, Ws[l], xs, NN, K);
    gemm_wmma_rows<<<gblocks, 128, 0, stream>>>(Hin, Wd[l], xd, NN, K);
    rowdot64<<<nb(NN, T), T, 0, stream>>>(xs, As[l], es, NN);
    rowdot64<<<nb(NN, T), T, 0, stream>>>(xd, Ad[l], edv, NN);
    fill_u32<<<nb(NN, T), T, 0, stream>>>(mbuf, 0x007FFFFFu /* mono(-inf) */, NN);
    fill_f32<<<nb(NN, T), T, 0, stream>>>(zbuf, 0.f, NN);
    fill_f32<<<nb((long)NN * DD, T), T, 0, stream>>>(acc, 0.f, NN * DD);
    edge_logits<<<nb(EE, T), T, 0, stream>>>(es, edv, src, dst, ebuf, mbuf, EE);
    edge_expsum<<<nb(EE, T), T, 0, stream>>>(ebuf, mbuf, dst, zbuf, EE);
    edge_scatter<<<nb((long)EE * 16, T), T, 0, stream>>>(ebuf, zbuf, src, dst, xs, acc,
                                                         EE * 16);
    node_finalize<<<nb((long)NN * DD, T), T, 0, stream>>>(acc, Bs[l], h, NN * DD,
                                                          (l < 2) ? 1 : 0);
    Hin = h;
    K = DD;
  }

  fill_f32<<<nb(GG * DD, T), T, 0, stream>>>(gsum, 0.f, GG * DD);
  fill_f32<<<1, T, 0, stream>>>(gcnt, 0.f, GG);
  pool_cnt<<<nb(NN, T), T, 0, stream>>>(batch, gcnt, NN);
  pool_sum<<<nb((long)NN * DD, T), T, 0, stream>>>(h, batch, gsum, NN * DD);
  final_linear<<<1, 256, 0, stream>>>(gsum, gcnt, post, lw, lb, (float*)d_out);
}